// gnn_21861383536723
// MI455X (gfx1250) — compile-verified
//
#include <hip/hip_runtime.h>
#include <hip/hip_bf16.h>

typedef __attribute__((ext_vector_type(2))) float v2f;
typedef __attribute__((ext_vector_type(8))) float v8f;

// ---------------------------------------------------------------------------
// Stage 0: zero the segment-sum accumulator (ws is poisoned, must re-zero
// every call; deterministic).
// ---------------------------------------------------------------------------
__global__ void gnn_zero_kernel(float4* __restrict__ a4, int n4) {
    int i = blockIdx.x * blockDim.x + threadIdx.x;
    if (i < n4) a4[i] = make_float4(0.f, 0.f, 0.f, 0.f);
}

// ---------------------------------------------------------------------------
// Stage 1: m = sc * tc, scatter-add onto destination nodes.
// 233 MB streamed once; accumulator (6.4 MB) is L2-resident, so the
// global_atomic_add_f32 traffic never touches HBM.
// ---------------------------------------------------------------------------
__global__ void gnn_edge_kernel(const float* __restrict__ tc,
                                const float* __restrict__ sc,
                                const int*   __restrict__ edge_dst,
                                float*       __restrict__ a,
                                int E) {
    int e = blockIdx.x * blockDim.x + threadIdx.x;
    if (e >= E) return;
    float s = sc[e];
    int   d = edge_dst[e];
    const float4* t4 = (const float4*)(tc + (size_t)e * 16);
    float* ad = a + (size_t)d * 16;
#pragma unroll
    for (int q = 0; q < 4; ++q) {
        float4 v = t4[q];
        atomicAdd(ad + 4 * q + 0, s * v.x);
        atomicAdd(ad + 4 * q + 1, s * v.y);
        atomicAdd(ad + 4 * q + 2, s * v.z);
        atomicAdd(ad + 4 * q + 3, s * v.w);
    }
}

// ---------------------------------------------------------------------------
// Stage 2: per 16-node tile:  g = relu(a @ Wg^T + bg)           [16x128]
//                             h2 = g @ W2^T + b2                 [16x32]
//                             z  = h2 @ W3^T + b3 -> softmax     [16x4]
// All GEMMs on V_WMMA_F32_16X16X4_F32 (full f32 precision, K chained in 4s).
// One wave per tile; 8 waves / block.
// ---------------------------------------------------------------------------
#define SGW_STRIDE 18     // gcn_w   [128][16] padded   (bank-conflict free)
#define SF2_STRIDE 130    // fc2_w   [32][128] padded
#define SF3_STRIDE 34     // fc3_w   [16][32]  padded, rows 4..15 zeroed
#define GT_STRIDE  20     // per-wave g bounce tile [16][16] padded
#define H2_STRIDE  36     // per-wave h2 tile       [16][32] padded

__launch_bounds__(256)
__global__ void gnn_node_kernel(const float* __restrict__ a,
                                const float* __restrict__ gcn_w,
                                const float* __restrict__ gcn_b,
                                const float* __restrict__ fc2_w,
                                const float* __restrict__ fc2_b,
                                const float* __restrict__ fc3_w,
                                const float* __restrict__ fc3_b,
                                float*       __restrict__ out,
                                int ntiles, int nnodes) {
    __shared__ float s_gcnw[128 * SGW_STRIDE];
    __shared__ float s_fc2w[32 * SF2_STRIDE];
    __shared__ float s_fc3w[16 * SF3_STRIDE];
    __shared__ float s_gcnb[128];
    __shared__ float s_fc2b[32];
    __shared__ float s_fc3b[16];
    __shared__ float s_gt[8][16 * GT_STRIDE];
    __shared__ float s_h2[8][16 * H2_STRIDE];
    __shared__ float s_z[8][16 * 4];

    const int tid = threadIdx.x;

    // ---- stage weights into LDS (once per block) ----
    for (int i = tid; i < 128 * 16; i += 256)
        s_gcnw[(i >> 4) * SGW_STRIDE + (i & 15)] = gcn_w[i];
    for (int i = tid; i < 32 * 128; i += 256)
        s_fc2w[(i >> 7) * SF2_STRIDE + (i & 127)] = fc2_w[i];
    for (int i = tid; i < 16 * SF3_STRIDE; i += 256)
        s_fc3w[i] = 0.f;
    if (tid < 128) s_gcnb[tid] = gcn_b[tid];
    if (tid < 32)  s_fc2b[tid] = fc2_b[tid];
    if (tid < 16)  s_fc3b[tid] = 0.f;
    __syncthreads();
    if (tid < 4 * 32)
        s_fc3w[(tid >> 5) * SF3_STRIDE + (tid & 31)] = fc3_w[tid];
    if (tid < 4) s_fc3b[tid] = fc3_b[tid];
    __syncthreads();

    const int wave = tid >> 5;
    const int lane = tid & 31;
    const int tile = blockIdx.x * 8 + wave;
    if (tile >= ntiles) return;

    const int half = lane >> 4;   // K-half selector in A/B fragments
    const int ln   = lane & 15;   // M (A) / N (B,C,D) index
    const int rowBase = tile * 16;

    // ---- A fragments for the gcn GEMM: a[16x16] tile, K chunks of 4 ----
    const float* arow = a + (size_t)(rowBase + ln) * 16;
    v2f afr[4];
#pragma unroll
    for (int c = 0; c < 4; ++c) {
        float2 v = *(const float2*)(arow + 4 * c + 2 * half);
        afr[c] = (v2f){v.x, v.y};
    }

    v8f acc2_0 = {};   // fc2 output cols 0..15, accumulated over all K
    v8f acc2_1 = {};   // fc2 output cols 16..31

    for (int t = 0; t < 8; ++t) {        // 8 column-tiles of g (128 = 8*16)
        // g_t = a @ Wg^T (cols 16t..16t+15), K = 16 in 4 chained WMMAs
        v8f g = {};
#pragma unroll
        for (int c = 0; c < 4; ++c) {
            float2 w = *(const float2*)&s_gcnw[(t * 16 + ln) * SGW_STRIDE + 4 * c + 2 * half];
            v2f bf = (v2f){w.x, w.y};
            g = __builtin_amdgcn_wmma_f32_16x16x4_f32(false, afr[c], false, bf,
                                                      (short)0, g, false, false);
        }
        // bias + relu, bounce through LDS to re-layout D (col-per-lane)
        // into A fragments (row-per-lane) for the fc2 accumulation.
        float bias = s_gcnb[t * 16 + ln];
#pragma unroll
        for (int j = 0; j < 8; ++j) {
            float gv = g[j] + bias;
            gv = gv > 0.f ? gv : 0.f;
            s_gt[wave][(half * 8 + j) * GT_STRIDE + ln] = gv;
        }
        asm volatile("s_wait_dscnt 0" ::: "memory");
        // fc2 partial: this 16-wide K chunk feeds both output col-tiles
#pragma unroll
        for (int c = 0; c < 4; ++c) {
            float2 ga = *(const float2*)&s_gt[wave][ln * GT_STRIDE + 4 * c + 2 * half];
            v2f af = (v2f){ga.x, ga.y};
            float2 w0 = *(const float2*)&s_fc2w[ln * SF2_STRIDE + t * 16 + 4 * c + 2 * half];
            float2 w1 = *(const float2*)&s_fc2w[(16 + ln) * SF2_STRIDE + t * 16 + 4 * c + 2 * half];
            v2f b0 = (v2f){w0.x, w0.y};
            v2f b1 = (v2f){w1.x, w1.y};
            acc2_0 = __builtin_amdgcn_wmma_f32_16x16x4_f32(false, af, false, b0,
                                                           (short)0, acc2_0, false, false);
            acc2_1 = __builtin_amdgcn_wmma_f32_16x16x4_f32(false, af, false, b1,
                                                           (short)0, acc2_1, false, false);
        }
        asm volatile("" ::: "memory");   // keep loads ordered before next t's stores
    }

    // ---- h2 = acc2 + b2, stage to LDS for the fc3 A fragments ----
    {
        float b0 = s_fc2b[ln];
        float b1 = s_fc2b[16 + ln];
#pragma unroll
        for (int j = 0; j < 8; ++j) {
            s_h2[wave][(half * 8 + j) * H2_STRIDE + ln]      = acc2_0[j] + b0;
            s_h2[wave][(half * 8 + j) * H2_STRIDE + 16 + ln] = acc2_1[j] + b1;
        }
        asm volatile("s_wait_dscnt 0" ::: "memory");
    }

    // ---- fc3: [16x32] @ [32x16-padded] -> z in cols 0..3 ----
    v8f acc3 = {};
#pragma unroll
    for (int c = 0; c < 8; ++c) {
        float2 ha = *(const float2*)&s_h2[wave][ln * H2_STRIDE + 4 * c + 2 * half];
        v2f af = (v2f){ha.x, ha.y};
        float2 w = *(const float2*)&s_fc3w[ln * SF3_STRIDE + 4 * c + 2 * half];
        v2f bf = (v2f){w.x, w.y};
        acc3 = __builtin_amdgcn_wmma_f32_16x16x4_f32(false, af, false, bf,
                                                     (short)0, acc3, false, false);
    }
    if (ln < 4) {
        float b3 = s_fc3b[ln];
#pragma unroll
        for (int j = 0; j < 8; ++j)
            s_z[wave][(half * 8 + j) * 4 + ln] = acc3[j] + b3;
    }
    asm volatile("s_wait_dscnt 0" ::: "memory");

    // ---- softmax over 4 logits per row, one row per lane 0..15 ----
    if (lane < 16 && rowBase + lane < nnodes) {
        float z0 = s_z[wave][lane * 4 + 0];
        float z1 = s_z[wave][lane * 4 + 1];
        float z2 = s_z[wave][lane * 4 + 2];
        float z3 = s_z[wave][lane * 4 + 3];
        float m = fmaxf(fmaxf(z0, z1), fmaxf(z2, z3));
        float e0 = __expf(z0 - m), e1 = __expf(z1 - m);
        float e2 = __expf(z2 - m), e3 = __expf(z3 - m);
        float inv = 1.f / (e0 + e1 + e2 + e3);
        float4 o = make_float4(e0 * inv, e1 * inv, e2 * inv, e3 * inv);
        *(float4*)(out + (size_t)(rowBase + lane) * 4) = o;
    }
}

// ---------------------------------------------------------------------------
// Launch
// ---------------------------------------------------------------------------
extern "C" void kernel_launch(void* const* d_in, const int* in_sizes, int n_in,
                              void* d_out, int out_size, void* d_ws, size_t ws_size,
                              hipStream_t stream) {
    // setup_inputs() order:
    // 0:x 1:feat 2:tc 3:sc 4:edge_dst 5:fc1_w 6:fc1_b 7:fcfeat1_w 8:fcfeat1_b
    // 9:gcn_w 10:gcn_b 11:fc2_w 12:fc2_b 13:fc3_w 14:fc3_b
    const float* tc       = (const float*)d_in[2];
    const float* sc       = (const float*)d_in[3];
    const int*   edge_dst = (const int*)d_in[4];
    const float* gcn_w    = (const float*)d_in[9];
    const float* gcn_b    = (const float*)d_in[10];
    const float* fc2_w    = (const float*)d_in[11];
    const float* fc2_b    = (const float*)d_in[12];
    const float* fc3_w    = (const float*)d_in[13];
    const float* fc3_b    = (const float*)d_in[14];
    float*       out      = (float*)d_out;

    const int N = in_sizes[0] / 4;   // x is [N,4]
    const int E = in_sizes[4];       // edge_dst is [E]
    const int ntiles = (N + 15) / 16;

    float* a = (float*)d_ws;         // [ntiles*16, 16] accumulator (6.4 MB)

    // Stage 0: zero accumulator (padded rows included -> safe WMMA reads)
    {
        int n4 = ntiles * 16 * 16 / 4;
        int blocks = (n4 + 255) / 256;
        gnn_zero_kernel<<<blocks, 256, 0, stream>>>((float4*)a, n4);
    }
    // Stage 1: edge scatter-sum
    {
        int blocks = (E + 255) / 256;
        gnn_edge_kernel<<<blocks, 256, 0, stream>>>(tc, sc, edge_dst, a, E);
    }
    // Stage 2: node MLP + softmax (WMMA)
    {
        int blocks = (ntiles + 7) / 8;
        gnn_node_kernel<<<blocks, 256, 0, stream>>>(a, gcn_w, gcn_b, fc2_w, fc2_b,
                                                    fc3_w, fc3_b, out, ntiles, N);
    }
}